// ReassemblePatchesLayer_39015482917582
// MI455X (gfx1250) — compile-verified
//
#include <hip/hip_runtime.h>
#include <hip/hip_bf16.h>
#include <stdint.h>

// Problem constants (match the reference module)
#define BB     512
#define NN     64        // patch size
#define CC     4         // channels (GRIDSIZE^2)
#define PP     192       // padded canvas
#define PADLO  64        // (PP - NN) / 2

// LDS tile: zero-bordered 66 x 66 x 4 (channel-innermost, matches global layout)
#define LROWS      66
#define LCOLS      66
#define LDS_FLOATS (LROWS * LCOLS * CC)     // 17424 floats = 69,696 B

#if defined(__has_builtin)
#if __has_builtin(__builtin_amdgcn_tensor_load_to_lds) && \
    __has_builtin(__builtin_amdgcn_s_wait_tensorcnt)
#define USE_TDM 1
#endif
#endif
#ifndef USE_TDM
#define USE_TDM 0
#endif

typedef unsigned int v4u __attribute__((ext_vector_type(4)));
typedef int          v8i __attribute__((ext_vector_type(8)));
typedef int          v4i __attribute__((ext_vector_type(4)));

__global__ __launch_bounds__(256) void
ReassemblePatchesLayer_39015482917582_kernel(const float* __restrict__ patches,
                                             const float* __restrict__ positions,
                                             float* __restrict__ out) {
    extern __shared__ float lds[];          // LDS_FLOATS floats
    const int b   = blockIdx.x;
    const int tid = threadIdx.x;

    // ---- 1) zero the whole LDS tile (gives the zero halo for free taps) ----
    float4* lds4 = reinterpret_cast<float4*>(lds);
    for (int i = tid; i < LDS_FLOATS / 4; i += 256) {
        lds4[i] = make_float4(0.f, 0.f, 0.f, 0.f);
    }
    __syncthreads();   // DS zero-stores drained before the DMA writes interior

    // ---- 2) DMA-stage the 64x64x4 patch (64 KB) into the tile interior ----
    const float* src = patches + (size_t)b * (NN * NN * CC);

#if USE_TDM
    // One TDM descriptor per block: 2D tile, 64 rows x 256 DWORDs, with LDS
    // padding of 8 DWORDs after every 256 DWORDs -> row pitch 264 floats
    // (= 66 texels x 4 ch), landing at interior texel (1,1).
    // Wave-uniform scalar guard: TDM ignores EXEC, so waves 1..7 must BRANCH
    // around the instruction, not merely be masked off.
    if (__builtin_amdgcn_readfirstlane((unsigned)tid >> 5) == 0u) {
        const uint64_t ga = (uint64_t)(uintptr_t)src;
        const uint32_t la = (uint32_t)(uintptr_t)&lds[(LCOLS + 1) * CC];

        v4u g0;
        g0.x = 1u;                                   // count=1, user mode, no gather
        g0.y = la;                                   // lds_addr (bytes)
        g0.z = (uint32_t)ga;                         // global_addr[31:0]
        g0.w = (uint32_t)((ga >> 32) & 0x01FFFFFFu)  // global_addr[56:32]
               | (2u << 30);                         // type = 2 ("image")

        v8i g1;
        g1[0] = (int)((2u << 16)       // data_size = 2 -> 4-byte elements
                    | (1u << 20)       // pad_enable
                    | (7u << 22)       // pad_interval: 7 -> every 256 DWORDs
                    | (7u << 25));     // pad_amount:   7 -> 8 DWORDs (pitch gap)
        g1[1] = (int)(256u << 16);     // tensor_dim0[15:0]=256 @ [63:48]
        g1[2] = (int)( 64u << 16);     // tensor_dim0 hi=0; tensor_dim1[15:0]=64 @ [95:80]
        g1[3] = (int)(256u << 16);     // tensor_dim1 hi=0; tile_dim0=256 @ [127:112]
        g1[4] = (int)  64u;            // tile_dim1=64 @ [143:128]; tile_dim2=0
        g1[5] = (int) 256u;            // tensor_dim0_stride lo32 = 256
        g1[6] = 0;                     // stride hi16 = 0; tensor_dim1_stride lo16 = 0
        g1[7] = 0;                     // tensor_dim1_stride hi = 0

        const v4i gz4 = {0, 0, 0, 0};              // groups 2/3 unused (<=2D)
        const v8i gz8 = {0, 0, 0, 0, 0, 0, 0, 0};  // trailing operand (clang-23 form)

        __builtin_amdgcn_tensor_load_to_lds(g0, g1, gz4, gz4, gz8, 0);
        __builtin_amdgcn_s_wait_tensorcnt(0);
    }
#else
    // Fallback: per-lane async global->LDS b128 (one 16B texel = 4 channels)
    for (int i = tid; i < NN * NN; i += 256) {        // 4096/256 = 16 iters
        const int r = i >> 6;
        const int c = i & 63;
        const uint32_t ldsoff =
            (uint32_t)(uintptr_t)&lds[(((r + 1) * LCOLS) + (c + 1)) * CC];
        const float* g = src + (size_t)i * CC;
        asm volatile("global_load_async_to_lds_b128 %0, %1, off"
                     :: "v"(ldsoff), "v"(g)
                     : "memory");
    }
    asm volatile("s_wait_asynccnt 0x0" ::: "memory");
#endif
    __syncthreads();

    // ---- 3) per-channel offsets: positions[b,0,j,c] -> flat b*8 + j*4 + c --
    float dx[CC], dy[CC];
    const float* pb = positions + (size_t)b * (2 * CC);
    #pragma unroll
    for (int c = 0; c < CC; ++c) {
        dx[c] = pb[c];
        dy[c] = pb[CC + c];
    }

    // ---- 4) gather: every output pixel, channels early-out by support box --
    float* outb = out + (size_t)b * (PP * PP);
    const int GROUPS = PP * PP / 4;                      // 9216 float4 groups
    for (int g = tid; g < GROUPS; g += 256) {
        const int y  = g / (PP / 4);
        const int x0 = (g % (PP / 4)) * 4;
        float acc0 = 0.f, acc1 = 0.f, acc2 = 0.f, acc3 = 0.f;

        #pragma unroll
        for (int c = 0; c < CC; ++c) {
            // patch-space row coordinate; nonzero only for -1 < ry < 64
            const float ry = (float)y - dy[c] - (float)PADLO;
            if (ry <= -1.0f || ry >= (float)NN) continue;
            const float frf = floorf(ry);
            const int   r0  = (int)frf;                  // in [-1, 63]
            const float wy  = ry - frf;
            const float* row0 = &lds[((r0 + 1) * LCOLS) * CC + c];
            const float* row1 = row0 + LCOLS * CC;

            const float rxbase = (float)x0 - dx[c] - (float)PADLO;
            #pragma unroll
            for (int k = 0; k < 4; ++k) {
                const float rx = rxbase + (float)k;
                if (rx > -1.0f && rx < (float)NN) {
                    const float fcf = floorf(rx);
                    const int   c0  = (int)fcf;          // in [-1, 63]
                    const float wx  = rx - fcf;
                    const int   o   = (c0 + 1) * CC;
                    const float p00 = row0[o], p01 = row0[o + CC];
                    const float p10 = row1[o], p11 = row1[o + CC];
                    const float top = p00 + wx * (p01 - p00);
                    const float bot = p10 + wx * (p11 - p10);
                    const float v   = top + wy * (bot - top);
                    if      (k == 0) acc0 += v;
                    else if (k == 1) acc1 += v;
                    else if (k == 2) acc2 += v;
                    else             acc3 += v;
                }
            }
        }
        *reinterpret_cast<float4*>(outb + y * PP + x0) =
            make_float4(acc0, acc1, acc2, acc3);
    }
}

extern "C" void kernel_launch(void* const* d_in, const int* in_sizes, int n_in,
                              void* d_out, int out_size, void* d_ws, size_t ws_size,
                              hipStream_t stream) {
    const float* patches   = (const float*)d_in[0];   // (512, 64, 64, 4) f32
    const float* positions = (const float*)d_in[1];   // (512, 1, 2, 4)  f32
    float*       out       = (float*)d_out;           // (512, 192, 192, 1) f32

    const size_t lds_bytes = (size_t)LDS_FLOATS * sizeof(float); // 69,696 B
    dim3 grid(BB), block(256);
    ReassemblePatchesLayer_39015482917582_kernel<<<grid, block, lds_bytes, stream>>>(
        patches, positions, out);
}